// WeightedEmbeddingBag_14585708937399
// MI455X (gfx1250) — compile-verified
//
#include <hip/hip_runtime.h>

typedef __attribute__((ext_vector_type(2))) float v2f;
typedef __attribute__((ext_vector_type(8))) float v8f;

#define B_BATCH 4096
#define N_SEQ   200
#define M_BAGS  20
#define D_DIM   64

// One wave32 per batch. score[b] = S (20x200 selection*psw) x E (200x64 gathered rows),
// tiled as V_WMMA_F32_16X16X4_F32: 2 M-tiles x 4 D-tiles, 50 K-steps of 4.
// The embedding table (25.6 MB) is L2-resident (192 MB L2); the B-tile gather loads
// ARE the bandwidth-critical stream, WMMAs ride along for free.
__global__ __launch_bounds__(32)
void webag_wmma_kernel(const int* __restrict__ input,
                       const int* __restrict__ offsets,
                       const float* __restrict__ psw,
                       const float* __restrict__ weight,
                       float* __restrict__ out)
{
    __shared__ int   s_idxoff[N_SEQ];   // byte offset of each gathered row in `weight`
    __shared__ float s_psw[N_SEQ];
    __shared__ int   s_bag[N_SEQ];      // bag id per sequence position

    const int b    = blockIdx.x;
    const int lane = threadIdx.x;       // 0..31

    const int*   in_row  = input   + b * N_SEQ;
    const int*   off_row = offsets + b * M_BAGS;
    const float* psw_row = psw     + b * N_SEQ;

    // offsets[b,*] are wave-uniform -> scalar loads into registers
    int offs[M_BAGS];
    #pragma unroll
    for (int j = 0; j < M_BAGS; ++j) offs[j] = off_row[j];

    // Stage per-n data into LDS: row byte-offset, weight, bag id.
    // bag(n) = count of offsets[j] < n  (n past last offset -> bag 20, never stored)
    for (int n = lane; n < N_SEQ; n += 32) {
        int   idx = in_row[n];
        float w   = psw_row[n];
        int   bag = 0;
        #pragma unroll
        for (int j = 0; j < M_BAGS; ++j) bag += (offs[j] < n) ? 1 : 0;
        s_idxoff[n] = idx * (D_DIM * 4);
        s_psw[n]    = w;
        s_bag[n]    = bag;
    }
    __syncthreads();

    const int half = lane >> 4;   // 0 -> K pair {0,1}; 1 -> K pair {2,3}
    const int col  = lane & 15;   // A: M row in tile; B: N column in tile

    v8f acc0[4], acc1[4];
    #pragma unroll
    for (int t = 0; t < 4; ++t) {
        acc0[t] = (v8f)0.0f;
        acc1[t] = (v8f)0.0f;
    }

    const char* wbytes = (const char*)weight;

    for (int n0 = 0; n0 < N_SEQ; n0 += 4) {
        const int nA = n0 + half * 2;

        // pair loads from LDS (consecutive n, 8B-aligned -> ds_load_2addr_b64)
        float p0 = s_psw[nA],    p1 = s_psw[nA + 1];
        int   g0 = s_bag[nA],    g1 = s_bag[nA + 1];
        int   o0 = s_idxoff[nA], o1 = s_idxoff[nA + 1];

        // A tiles: selection * per-sample-weight
        v2f a0, a1;
        a0.x = (g0 == col)      ? p0 : 0.0f;
        a0.y = (g1 == col)      ? p1 : 0.0f;
        a1.x = (g0 == col + 16) ? p0 : 0.0f;
        a1.y = (g1 == col + 16) ? p1 : 0.0f;

        // B tiles: the embedding gather itself (half-wave reads 64B contiguous per row)
        const float* r0 = (const float*)(wbytes + o0);
        const float* r1 = (const float*)(wbytes + o1);

        #pragma unroll
        for (int dt = 0; dt < 4; ++dt) {
            v2f bt;
            bt.x = r0[dt * 16 + col];
            bt.y = r1[dt * 16 + col];
            acc0[dt] = __builtin_amdgcn_wmma_f32_16x16x4_f32(
                false, a0, false, bt, (short)0, acc0[dt], false, false);
            acc1[dt] = __builtin_amdgcn_wmma_f32_16x16x4_f32(
                false, a1, false, bt, (short)0, acc1[dt], false, false);
        }
    }

    // C/D layout: VGPR r -> M = 8*half + r (tile 0), 16 + 8*half + r (tile 1); N = col.
    float* out_b = out + (size_t)b * (M_BAGS * D_DIM);

    // Tile 0: all 16 rows valid -> 32 unconditional stores (clean store clauses).
    #pragma unroll
    for (int dt = 0; dt < 4; ++dt) {
        #pragma unroll
        for (int r = 0; r < 8; ++r) {
            int m0 = half * 8 + r;
            out_b[m0 * D_DIM + dt * 16 + col] = acc0[dt][r];
        }
    }

    // Tile 1: only rows 16..19 are real bags -> half==0 lanes, r<4 (compile-time).
    if (half == 0) {
        #pragma unroll
        for (int dt = 0; dt < 4; ++dt) {
            #pragma unroll
            for (int r = 0; r < 4; ++r) {
                out_b[(16 + r) * D_DIM + dt * 16 + col] = acc1[dt][r];
            }
        }
    }
}

extern "C" void kernel_launch(void* const* d_in, const int* in_sizes, int n_in,
                              void* d_out, int out_size, void* d_ws, size_t ws_size,
                              hipStream_t stream) {
    const int*   input   = (const int*)d_in[0];
    const int*   offsets = (const int*)d_in[1];
    const float* pswp    = (const float*)d_in[2];
    const float* weight  = (const float*)d_in[3];
    float*       out     = (float*)d_out;

    webag_wmma_kernel<<<B_BATCH, 32, 0, stream>>>(input, offsets, pswp, weight, out);
}